// MoEAttention_22239340658920
// MI455X (gfx1250) — compile-verified
//
#include <hip/hip_runtime.h>
#include <math.h>
#include <stdint.h>

// ---------------------------------------------------------------------------
// Problem constants (match reference)
// ---------------------------------------------------------------------------
#define HIDDEN  2048
#define NHEADS  16
#define NKV     4
#define HEADDIM 128
#define BATCH   2
#define SEQ     2048
#define KC      16            // GEMM K-chunk staged in LDS per step

typedef __attribute__((ext_vector_type(2))) float v2f;
typedef __attribute__((ext_vector_type(8))) float v8f;

// CDNA5 fp32 WMMA: D(16x16,f32) = A(16x4,f32) * B(4x16,f32) + C
// A frag (2 VGPRs): lanes 0-15 -> M=lane, {K=0,K=1}; lanes 16-31 -> M=lane-16, {K=2,K=3}
// B frag (2 VGPRs): lanes 0-15 -> N=lane, {K=0,K=1}; lanes 16-31 -> N=lane-16, {K=2,K=3}
// C/D (8 VGPRs): vgpr j, lanes 0-15 -> (M=j, N=lane); lanes 16-31 -> (M=j+8, N=lane-16)
__device__ __forceinline__ v8f wmma_f32(v2f a, v2f b, v8f c) {
  return __builtin_amdgcn_wmma_f32_16x16x4_f32(
      /*neg_a=*/false, a, /*neg_b=*/false, b,
      /*c_mod=*/(short)0, c, /*reuse_a=*/false, /*reuse_b=*/false);
}

// ---------------------------------------------------------------------------
// fp32 WMMA GEMM with async-to-LDS double-buffered B staging.
//   C[M,N] = A[M,K] * B[K,N], all row-major.
// Block = 256 threads (8 waves); block tile = 256(M) x 64(N).
// Wave w computes rows [row0 + 32w, +32) x 64 cols  (8 v8f accumulators).
// B is staged in KCx64 fp32 chunks via GLOBAL_LOAD_ASYNC_TO_LDS_B128
// (ASYNCcnt), double-buffered, shared by all 8 waves (8x L2-traffic cut).
// The LDS destination address is derived from &Btile (low 32 bits of the
// generic pointer == DS offset on gfx1250); the ptrtoint capture makes the
// asm "memory" clobber cover Btile so the consumer ds_loads are kept.
// mode 0: store row-major [M,N];  mode 1: store reshaped [B, heads, S, D].
// ---------------------------------------------------------------------------
__global__ __launch_bounds__(256)
void gemm16(const float* __restrict__ A, const float* __restrict__ B,
            float* __restrict__ C, int M, int N, int K, int mode, int heads) {
  __shared__ __align__(16) float Btile[2][KC][64];

  const int tid  = threadIdx.x;
  const int lane = tid & 31;
  const int half = lane >> 4;
  const int l16  = lane & 15;
  const int wave = tid >> 5;
  const int row0 = blockIdx.y * 256 + wave * 32;
  const int col0 = blockIdx.x * 64;

  // --- async-copy mapping: 256 threads x 16B = one KCx64 fp32 chunk --------
  const int lr = tid >> 4;            // chunk row 0..15
  const int lc = (tid & 15) * 4;      // 4-float column group
  const float* gsrc = B + (size_t)lr * N + col0 + lc;
  // LDS byte offset of this thread's 16B slot in buffer 0 (captures Btile)
  const unsigned lds0 = (unsigned)(uintptr_t)&Btile[0][lr][lc];

  // prologue: stage chunk 0 into buffer 0
  asm volatile("global_load_async_to_lds_b128 %0, %1, off"
               :: "v"(lds0), "v"(gsrc) : "memory");

  const float* arow0 = A + (size_t)(row0 + l16) * K + 2 * half;
  const float* arow1 = A + (size_t)(row0 + 16 + l16) * K + 2 * half;

  v8f acc[8] = {};                    // [rt*4 + ct]

  const int nchunks = K / KC;
  for (int ch = 0; ch < nchunks; ++ch) {
    const int buf = ch & 1;
    if (ch + 1 < nchunks) {
      // stage next chunk into the other buffer, then wait for current chunk
      const float*   nsrc = gsrc + (size_t)(ch + 1) * KC * N;
      const unsigned nlds = lds0 + (unsigned)((ch + 1) & 1) * (KC * 64 * 4u);
      asm volatile("global_load_async_to_lds_b128 %0, %1, off"
                   :: "v"(nlds), "v"(nsrc) : "memory");
      asm volatile("s_wait_asynccnt 0x1" ::: "memory");
    } else {
      asm volatile("s_wait_asynccnt 0x0" ::: "memory");
    }
    __syncthreads();                  // chunk `ch` fully visible to all waves

    #pragma unroll
    for (int kk = 0; kk < KC; kk += 4) {
      const int k0 = ch * KC + kk;
      v2f a0, a1;
      a0.x = arow0[k0];  a0.y = arow0[k0 + 1];
      a1.x = arow1[k0];  a1.y = arow1[k0 + 1];
      #pragma unroll
      for (int t = 0; t < 4; ++t) {
        v2f b;
        b.x = Btile[buf][kk + 2 * half][t * 16 + l16];
        b.y = Btile[buf][kk + 2 * half + 1][t * 16 + l16];
        acc[t]     = wmma_f32(a0, b, acc[t]);
        acc[4 + t] = wmma_f32(a1, b, acc[4 + t]);
      }
    }
    __syncthreads();                  // done reading buf before it is refilled
  }

  #pragma unroll
  for (int rt = 0; rt < 2; ++rt) {
    #pragma unroll
    for (int t = 0; t < 4; ++t) {
      v8f a = acc[rt * 4 + t];
      #pragma unroll
      for (int j = 0; j < 8; ++j) {
        const int m = row0 + rt * 16 + j + 8 * half;
        const int n = col0 + t * 16 + l16;
        size_t idx;
        if (mode == 0) {
          idx = (size_t)m * N + n;
        } else {
          const int bb = m / SEQ, s = m % SEQ;
          const int hh = n / HEADDIM, d = n % HEADDIM;
          idx = (((size_t)bb * heads + hh) * SEQ + s) * HEADDIM + d;
        }
        C[idx] = a[j];
      }
    }
  }
}

// ---------------------------------------------------------------------------
// RoPE (in place) on X laid out as [TH, S, D].  One thread handles one (d,d+64)
// pair:  x'[d]    = x[d]*cos - x[d+64]*sin
//        x'[d+64] = x[d+64]*cos + x[d]*sin,  angle = s * theta^(-d/64)
// ---------------------------------------------------------------------------
__global__ __launch_bounds__(256)
void rope_kernel(float* __restrict__ X, int total) {
  const int idx = blockIdx.x * blockDim.x + threadIdx.x;
  if (idx >= total) return;
  const int d   = idx & 63;
  const int row = idx >> 6;            // th*S + s
  const int s   = row & (SEQ - 1);
  const size_t base = (size_t)row * HEADDIM;
  const float inv = powf(10000.0f, -(float)d * (1.0f / 64.0f));
  const float ang = (float)s * inv;
  float sn, cs;
  sincosf(ang, &sn, &cs);
  const float x0 = X[base + d];
  const float x1 = X[base + d + 64];
  X[base + d]      = x0 * cs - x1 * sn;
  X[base + d + 64] = x1 * cs + x0 * sn;
}

// ---------------------------------------------------------------------------
// Flash attention, one wave per (b, head, 16-row query tile).
//   Q: [B, NH,  S, D]   K,V: [B, NKV, S, D]   mask: [S, S]   Ctx: [B, S, NH*D]
// Online softmax: row stats (m, l) replicated across the 16 lanes of each
// wave half (C-layout row j lives in vgpr j of one half).  P (16x16) is
// transposed C-layout -> A-layout through per-wave LDS scratch.
// K/V per (b,kv_head) is 1 MB -> L2-resident across the 128 q-tiles reusing it.
// ---------------------------------------------------------------------------
__global__ __launch_bounds__(256)
void attn_kernel(const float* __restrict__ Q, const float* __restrict__ Kb,
                 const float* __restrict__ Vb, const float* __restrict__ mask,
                 float* __restrict__ Ctx) {
  __shared__ float lds[8][16][17];   // per-wave 16x16 P scratch, padded
  const int lane = threadIdx.x & 31;
  const int half = lane >> 4;
  const int l16  = lane & 15;
  const int wave = threadIdx.x >> 5;

  const int wid = blockIdx.x * 8 + wave;
  const int qt  = wid & (SEQ / 16 - 1);     // 128 q-tiles
  const int h   = (wid >> 7) & (NHEADS - 1);
  const int b   = wid >> 11;
  const int kvh = h >> 2;                   // GQA: 4 q-heads per kv-head

  const float* Qbase = Q  + (((size_t)b * NHEADS + h) * SEQ + (size_t)qt * 16) * HEADDIM;
  const float* Kbase = Kb + (((size_t)b * NKV + kvh) * SEQ) * HEADDIM;
  const float* Vbase = Vb + (((size_t)b * NKV + kvh) * SEQ) * HEADDIM;
  const float* Mbase = mask + (size_t)(qt * 16) * SEQ;

  // Q tile as 32 A-fragments over the D=128 contraction dim
  v2f qa[32];
  #pragma unroll
  for (int s2 = 0; s2 < 32; ++s2) {
    const float* qp = Qbase + (size_t)l16 * HEADDIM + 4 * s2 + 2 * half;
    qa[s2].x = qp[0];
    qa[s2].y = qp[1];
  }

  v8f oacc[8] = {};               // 16 rows x 128 cols output accumulator
  float mr[8], lr[8];
  #pragma unroll
  for (int j = 0; j < 8; ++j) { mr[j] = -1e30f; lr[j] = 0.0f; }

  const float scale = 0.08838834764831845f;   // 1/sqrt(128)

  for (int kt = 0; kt < SEQ / 16; ++kt) {
    // ---- scores = Q * K^T : 32 fp32 WMMAs -----------------------------
    v8f sc = {};
    const float* kp = Kbase + ((size_t)kt * 16 + l16) * HEADDIM + 2 * half;
    #pragma unroll
    for (int s2 = 0; s2 < 32; ++s2) {
      v2f kb;
      kb.x = kp[4 * s2];
      kb.y = kp[4 * s2 + 1];
      sc = wmma_f32(qa[s2], kb, sc);
    }

    // ---- online softmax over the 16-key tile --------------------------
    float p[8];
    #pragma unroll
    for (int j = 0; j < 8; ++j) {
      float v = sc[j] * scale +
                Mbase[(size_t)(j + 8 * half) * SEQ + kt * 16 + l16];
      float rmax = v;
      rmax = fmaxf(rmax, __shfl_xor(rmax, 1, 16));
      rmax = fmaxf(rmax, __shfl_xor(rmax, 2, 16));
      rmax = fmaxf(rmax, __shfl_xor(rmax, 4, 16));
      rmax = fmaxf(rmax, __shfl_xor(rmax, 8, 16));
      const float mnew = fmaxf(mr[j], rmax);
      const float pe   = __expf(v - mnew);
      float rsum = pe;
      rsum += __shfl_xor(rsum, 1, 16);
      rsum += __shfl_xor(rsum, 2, 16);
      rsum += __shfl_xor(rsum, 4, 16);
      rsum += __shfl_xor(rsum, 8, 16);
      const float alpha = __expf(mr[j] - mnew);
      lr[j] = lr[j] * alpha + rsum;
      mr[j] = mnew;
      p[j]  = pe;
      #pragma unroll
      for (int t = 0; t < 8; ++t) oacc[t][j] *= alpha;
    }

    // ---- transpose P (C layout -> A layout) via per-wave LDS ----------
    #pragma unroll
    for (int j = 0; j < 8; ++j) lds[wave][j + 8 * half][l16] = p[j];
    asm volatile("s_wait_dscnt 0" ::: "memory");
    v2f pa[4];
    #pragma unroll
    for (int s2 = 0; s2 < 4; ++s2) {
      pa[s2].x = lds[wave][l16][4 * s2 + 2 * half];
      pa[s2].y = lds[wave][l16][4 * s2 + 2 * half + 1];
    }
    asm volatile("s_wait_dscnt 0" ::: "memory");

    // ---- out += P * V : 32 fp32 WMMAs ---------------------------------
    #pragma unroll
    for (int s2 = 0; s2 < 4; ++s2) {
      const float* vp = Vbase + ((size_t)kt * 16 + 4 * s2 + 2 * half) * HEADDIM + l16;
      #pragma unroll
      for (int t = 0; t < 8; ++t) {
        v2f vb;
        vb.x = vp[t * 16];
        vb.y = vp[HEADDIM + t * 16];
        oacc[t] = wmma_f32(pa[s2], vb, oacc[t]);
      }
    }
  }

  // ---- normalize and store into Ctx[B, S, NH*D] -----------------------
  #pragma unroll
  for (int t = 0; t < 8; ++t) {
    #pragma unroll
    for (int j = 0; j < 8; ++j) {
      const int sq = qt * 16 + j + 8 * half;
      Ctx[((size_t)b * SEQ + sq) * (NHEADS * HEADDIM) + h * HEADDIM + t * 16 + l16] =
          oacc[t][j] / lr[j];
    }
  }
}

// ---------------------------------------------------------------------------
// Launch
// ---------------------------------------------------------------------------
extern "C" void kernel_launch(void* const* d_in, const int* in_sizes, int n_in,
                              void* d_out, int out_size, void* d_ws, size_t ws_size,
                              hipStream_t stream) {
  const float* hidden = (const float*)d_in[0];
  const float* mask   = (const float*)d_in[1];
  const float* Wq     = (const float*)d_in[2];
  const float* Wk     = (const float*)d_in[3];
  const float* Wv     = (const float*)d_in[4];
  const float* Wo     = (const float*)d_in[5];
  float* out = (float*)d_out;

  // Workspace: Q [B,NH,S,D] | K [B,NKV,S,D] | V [B,NKV,S,D] | Ctx [B,S,NH*D]
  float* Qb  = (float*)d_ws;
  float* Kb  = Qb + (size_t)BATCH * NHEADS * SEQ * HEADDIM;
  float* Vb  = Kb + (size_t)BATCH * NKV * SEQ * HEADDIM;
  float* Ctx = Vb + (size_t)BATCH * NKV * SEQ * HEADDIM;

  const int M = BATCH * SEQ;                 // 4096
  dim3 blk(256);

  // QKV projections (WMMA fp32 + async-LDS B staging) -> [B, heads, S, D]
  gemm16<<<dim3(2048 / 64, M / 256), blk, 0, stream>>>(hidden, Wq, Qb, M, 2048, HIDDEN, 1, NHEADS);
  gemm16<<<dim3(512 / 64,  M / 256), blk, 0, stream>>>(hidden, Wk, Kb, M, 512,  HIDDEN, 1, NKV);
  gemm16<<<dim3(512 / 64,  M / 256), blk, 0, stream>>>(hidden, Wv, Vb, M, 512,  HIDDEN, 1, NKV);

  // RoPE on Q and K
  const int qtot = BATCH * NHEADS * SEQ * (HEADDIM / 2);
  const int ktot = BATCH * NKV    * SEQ * (HEADDIM / 2);
  rope_kernel<<<(qtot + 255) / 256, blk, 0, stream>>>(Qb, qtot);
  rope_kernel<<<(ktot + 255) / 256, blk, 0, stream>>>(Kb, ktot);

  // Flash attention: B*NH*(S/16) = 4096 waves, 8 waves per block
  const int nwaves = BATCH * NHEADS * (SEQ / 16);
  attn_kernel<<<nwaves / 8, blk, 0, stream>>>(Qb, Kb, Vb, mask, Ctx);

  // Output projection -> d_out row-major [B, S, HIDDEN]
  gemm16<<<dim3(2048 / 64, M / 256), blk, 0, stream>>>(Ctx, Wo, out, M, 2048, HIDDEN, 0, 1);
}